// TriangleLinear_8349416423506
// MI455X (gfx1250) — compile-verified
//
#include <hip/hip_runtime.h>

typedef float v2f __attribute__((ext_vector_type(2)));
typedef float v4f __attribute__((ext_vector_type(4)));
typedef float v8f __attribute__((ext_vector_type(8)));

#define N_IN   8192
#define N_OUT  8192
#define NB     8        // batch
#define KCH    1024     // split-K chunk
#define NCHUNK (N_IN / KCH)
#define NTILE  (N_OUT / 16)

// D(16x16) = A(16x4: M = 16 output rows of W) * B(4x16: N = batch cols) + C
// One wave per (16-row tile, K chunk). Lane l: M = N = l&15, K pair = 2*(l>>4).
// Batch columns 8..15 duplicate 0..7 (never stored) -> no A/B batch masking.
// Triangle masking only in a <=4-iteration ragged prefix (i < o0+12).
__global__ __launch_bounds__(32)
void tri_wmma_partial(const float* __restrict__ w, const float* __restrict__ x,
                      const float* __restrict__ bias, float* __restrict__ ws,
                      float* __restrict__ out, int k_chunk, int final_mode) {
  const int tile    = blockIdx.x;
  const int o0      = tile << 4;
  const int fc_min  = (o0 > 4) ? (o0 - 4) : 0;
  const int k_begin = blockIdx.y * k_chunk;
  int i_end = k_begin + k_chunk;
  if (i_end > N_IN) i_end = N_IN;
  if (i_end <= fc_min) return;                 // chunk entirely in the zero region
  int i = max(k_begin, fc_min & ~3);           // 4-aligned trimmed start

  const int lane = threadIdx.x;
  const int half = lane >> 4;                  // 0: K=0,1   1: K=2,3
  const int n    = lane & 15;
  const int k0   = half << 1;

  const int  o  = o0 + n;                      // W row carried by this lane (A)
  const int  fc = (o > 4) ? (o - 4) : 0;       // first stored column of row o
  const long rs = (long)o * N_IN - ((o >= 5) ? (((long)(o - 5) * (o - 4)) >> 1) : 0);
  const float* __restrict__ wrow = w + (rs - fc);      // wrow[i] == W[o][i], i >= fc
  const float* __restrict__ xrow = x + (long)(n & 7) * N_IN;  // dup batches 8..15

  v8f acc = {};

  // ---- ragged diagonal prefix: needs per-element masking (<=4 iterations) ----
  int rag_end = o0 + 12;                       // for i >= o0+12: i >= fc for all lanes
  if (rag_end > i_end) rag_end = i_end;
  for (; i < rag_end; i += 4) {
    const int ii = i + k0;
    float a0 = __builtin_nontemporal_load(wrow + ii);      // in-bounds even if ii<fc
    float a1 = __builtin_nontemporal_load(wrow + ii + 1);
    a0 = (ii     >= fc) ? a0 : 0.0f;
    a1 = (ii + 1 >= fc) ? a1 : 0.0f;
    v2f af; af.x = a0; af.y = a1;
    v2f bf = *(const v2f*)(xrow + ii);
    acc = __builtin_amdgcn_wmma_f32_16x16x4_f32(false, af, false, bf,
                                                (short)0, acc, false, false);
  }

  // ---- clean streaming loop: no masks, pure load+wmma ----
#pragma unroll 4
  for (; i < i_end; i += 4) {
    const int ii = i + k0;
    v2f af;
    af.x = __builtin_nontemporal_load(wrow + ii);          // single-use weight stream
    af.y = __builtin_nontemporal_load(wrow + ii + 1);
    v2f bf = *(const v2f*)(xrow + ii);                     // L2-resident x
    acc = __builtin_amdgcn_wmma_f32_16x16x4_f32(false, af, false, bf,
                                                (short)0, acc, false, false);
  }

  // D layout: lane l holds batch n=l&15, rows o0 + 8*(l>>4) + v (v = vgpr 0..7).
  if (n < NB) {
    const int obase = o0 + (half << 3);
    const v4f b0 = *(const v4f*)(bias + obase);            // rows obase..obase+3
    const v4f b1 = *(const v4f*)(bias + obase + 4);        // rows obase+4..obase+7
    if (final_mode) {
      // out[n][N_OUT-1-(obase+v)] is contiguous descending -> pack reversed.
      float* po = out + (long)n * N_OUT + (N_OUT - 8 - obase);  // address of v=7
      v4f lo, hi;
      lo.x = acc[7] + b1.w; lo.y = acc[6] + b1.z;
      lo.z = acc[5] + b1.y; lo.w = acc[4] + b1.x;
      hi.x = acc[3] + b0.w; hi.y = acc[2] + b0.z;
      hi.z = acc[1] + b0.y; hi.w = acc[0] + b0.x;
      *(v4f*)po       = lo;                                // 32B-aligned
      *(v4f*)(po + 4) = hi;
    } else {
      // ws[chunk][tile][batch][16 rows], rows index = 8*half + v
      float* p = ws + ((((long)blockIdx.y * NTILE + tile) * NB + n) << 4) + (half << 3);
      v4f lo, hi;
      lo.x = acc[0]; lo.y = acc[1]; lo.z = acc[2]; lo.w = acc[3];
      hi.x = acc[4]; hi.y = acc[5]; hi.z = acc[6]; hi.w = acc[7];
      *(v4f*)p       = lo;
      *(v4f*)(p + 4) = hi;
    }
  }
}

// Deterministic fixed-order reduction over K chunks + bias + flip.
__global__ __launch_bounds__(256)
void tri_reduce(const float* __restrict__ ws, const float* __restrict__ bias,
                float* __restrict__ out) {
  const int t = blockIdx.x * 256 + threadIdx.x;
  if (t >= N_OUT * NB) return;
  const int o = t >> 3, m = t & 7;
  const int tile = o >> 4, r = o & 15;
  const int fc_min = (((o & ~15) - 4) > 0) ? ((o & ~15) - 4) : 0;
  float s = 0.0f;
  for (int c = (fc_min / KCH); c < NCHUNK; ++c)            // only chunks stage 1 wrote
    s += ws[((((long)c * NTILE + tile) * NB + m) << 4) + r];
  out[(long)m * N_OUT + (N_OUT - 1 - o)] = s + bias[o];
}

extern "C" void kernel_launch(void* const* d_in, const int* in_sizes, int n_in,
                              void* d_out, int out_size, void* d_ws, size_t ws_size,
                              hipStream_t stream) {
  (void)in_sizes; (void)n_in; (void)out_size;
  const float* x    = (const float*)d_in[0];   // [8, 8192]
  const float* w    = (const float*)d_in[1];   // packed triangular weights
  const float* bias = (const float*)d_in[2];   // [8192]
  float*       out  = (float*)d_out;           // [8, 8192]

  const size_t ws_needed = (size_t)NCHUNK * NTILE * NB * 16 * sizeof(float);  // 2 MB
  if (ws_size >= ws_needed) {
    // Split-K for load balance across the triangle; two-stage deterministic sum.
    dim3 grid(NTILE, NCHUNK);
    tri_wmma_partial<<<grid, 32, 0, stream>>>(w, x, bias, (float*)d_ws, out,
                                              KCH, /*final=*/0);
    tri_reduce<<<(N_OUT * NB + 255) / 256, 256, 0, stream>>>(
        (const float*)d_ws, bias, out);
  } else {
    // Fallback: one wave per 16-row tile over full (trimmed) K.
    dim3 grid(NTILE, 1);
    tri_wmma_partial<<<grid, 32, 0, stream>>>(w, x, bias, nullptr, out,
                                              N_IN, /*final=*/1);
  }
}